// matchingA_81844896793233
// MI455X (gfx1250) — compile-verified
//
#include <hip/hip_runtime.h>
#include <hip/hip_bf16.h>
#include <math.h>

// Problem constants (match reference)
#define NSAMP 2048
#define SDIM  8
#define TDIM  64
#define CDIM  3
#define MID   30
#define PDIM  (CDIM * SDIM)    // 24 groups
#define KDIM  (PDIM * MID)     // 720 contraction length

typedef __attribute__((ext_vector_type(2))) float v2f;
typedef __attribute__((ext_vector_type(8))) float v8f;

// ---------------------------------------------------------------------------
// Kernel 1: per-sample projections u,v; cosine normalization; fold Wnorm/8
// into U. One block per sample a. Produces row-major U,V: (NSAMP, KDIM).
//   flat j-axis mapping (torch .view semantics): for e = p*MID + d,
//   p = s*CDIM + c, linear-output channel j = c*MID + d.
//   Atemp reshape groups p as (C,S) => weight index = p / SDIM.
// ---------------------------------------------------------------------------
__global__ __launch_bounds__(256)
void proj_norm_kernel(const float* __restrict__ x,
                      const float* __restrict__ Wu_w, const float* __restrict__ Wu_b,
                      const float* __restrict__ Wv_w, const float* __restrict__ Wv_b,
                      const float* __restrict__ Wnorm,
                      float* __restrict__ U, float* __restrict__ V)
{
    __shared__ float xs[SDIM * TDIM];     // 512 floats
    __shared__ float us[KDIM];            // 720
    __shared__ float vs[KDIM];            // 720
    __shared__ float scU[PDIM];           // per-group scale for U (incl. weight)
    __shared__ float scV[PDIM];           // per-group scale for V

    const int a = blockIdx.x;
    const int t = threadIdx.x;

    for (int i = t; i < SDIM * TDIM; i += 256)
        xs[i] = x[(size_t)a * SDIM * TDIM + i];
    __syncthreads();

    // u[e], v[e] for e in [0,720)
    for (int e = t; e < KDIM; e += 256) {
        const int p = e / MID, d = e % MID;
        const int s = p / CDIM, c = p % CDIM;
        const int j = c * MID + d;              // linear layer output channel
        const float* xr = &xs[s * TDIM];
        const float* wu = &Wu_w[(size_t)j * TDIM];
        const float* wv = &Wv_w[(size_t)j * TDIM];
        float au = Wu_b[j], av = Wv_b[j];
        #pragma unroll 8
        for (int k = 0; k < TDIM; ++k) {
            const float xv = xr[k];
            au = fmaf(xv, wu[k], au);
            av = fmaf(xv, wv[k], av);
        }
        us[e] = au;
        vs[e] = av;
    }
    __syncthreads();

    // group norms (24 groups of 30, for both u and v): 48 lanes
    if (t < 2 * PDIM) {
        const int p = t % PDIM;
        const bool isv = (t >= PDIM);
        const float* arr = isv ? vs : us;
        float ss = 0.0f;
        #pragma unroll
        for (int d = 0; d < MID; ++d) {
            const float val = arr[p * MID + d];
            ss = fmaf(val, val, ss);
        }
        const float sc = 1.0f / fmaxf(sqrtf(ss), 1e-8f);
        if (isv) scV[p] = sc;
        else     scU[p] = sc * Wnorm[p / SDIM] * (1.0f / SDIM);
    }
    __syncthreads();

    for (int e = t; e < KDIM; e += 256) {
        const int p = e / MID;
        U[(size_t)a * KDIM + e] = us[e] * scU[p];
        V[(size_t)a * KDIM + e] = vs[e] * scV[p];
    }
}

// ---------------------------------------------------------------------------
// Kernel 2: Anorm = U @ V^T  (2048 x 2048 x 720) via V_WMMA_F32_16X16X4_F32.
// One wave per 16x64 output strip (4 N-tiles share one A fragment).
// A fragment (16x4 f32, ISA layout): lane l holds A[l&15, 2*(l>>4)+{0,1}]
//   -> float2 load from row-major U.
// B fragment (4x16 f32, mirrored): lane l holds B[2*(l>>4)+{0,1}, l&15]
//   = V[col, k..k+1] -> float2 load from row-major V.
// ---------------------------------------------------------------------------
__global__ __launch_bounds__(256)
void gemm_wmma_kernel(const float* __restrict__ U, const float* __restrict__ V,
                      float* __restrict__ Anorm)
{
    const int wave   = (blockIdx.x * 256 + threadIdx.x) >> 5;
    const int lane   = threadIdx.x & 31;
    const int nGroup = wave & 31;          // 32 groups of 64 columns
    const int mTile  = wave >> 5;          // 128 tiles of 16 rows
    const int rowBase = mTile * 16;
    const int colBase = nGroup * 64;

    const int m  = lane & 15;              // also the B column-in-tile index
    const int kh = lane >> 4;              // K half select

    const float* aPtr = U + (size_t)(rowBase + m) * KDIM + 2 * kh;
    const float* bPtr = V + (size_t)(colBase + m) * KDIM + 2 * kh;

    v8f acc0 = {}, acc1 = {}, acc2 = {}, acc3 = {};

    for (int k0 = 0; k0 < KDIM; k0 += 4) {
        const v2f af = *(const v2f*)(aPtr + k0);
        const v2f b0 = *(const v2f*)(bPtr + k0);
        const v2f b1 = *(const v2f*)(bPtr + (size_t)16 * KDIM + k0);
        const v2f b2 = *(const v2f*)(bPtr + (size_t)32 * KDIM + k0);
        const v2f b3 = *(const v2f*)(bPtr + (size_t)48 * KDIM + k0);
        acc0 = __builtin_amdgcn_wmma_f32_16x16x4_f32(false, af, false, b0,
                                                     (short)0, acc0, false, false);
        acc1 = __builtin_amdgcn_wmma_f32_16x16x4_f32(false, af, false, b1,
                                                     (short)0, acc1, false, false);
        acc2 = __builtin_amdgcn_wmma_f32_16x16x4_f32(false, af, false, b2,
                                                     (short)0, acc2, false, false);
        acc3 = __builtin_amdgcn_wmma_f32_16x16x4_f32(false, af, false, b3,
                                                     (short)0, acc3, false, false);
    }

    // C/D layout: VGPR r -> row = rowBase + r + 8*(lane>>4), col = colBase + (lane&15)
    const int n = lane & 15;
    #pragma unroll
    for (int r = 0; r < 8; ++r) {
        const int row = rowBase + r + 8 * kh;
        float* orow = Anorm + (size_t)row * NSAMP + colBase + n;
        orow[0]  = acc0[r];
        orow[16] = acc1[r];
        orow[32] = acc2[r];
        orow[48] = acc3[r];
    }
}

// ---------------------------------------------------------------------------
// Kernel 3: out[i,j] = sigmoid(s*(A+B)[i,j] + (1-s)*(A+B)[j,i]) * (i != j)
// 16x16 tiles; transposed operands staged through LDS for coalescing.
// ---------------------------------------------------------------------------
__global__ __launch_bounds__(256)
void epilogue_kernel(const float* __restrict__ Anorm, const float* __restrict__ Bias,
                     const float* __restrict__ scalar_a, float* __restrict__ out)
{
    __shared__ float tT[16][17];           // (A+B) tile at (j0, i0), padded

    const int j0 = (blockIdx.x & (NSAMP / 16 - 1)) * 16;
    const int i0 = (blockIdx.x >> 7) * 16;
    const int tx = threadIdx.x & 15;
    const int ty = threadIdx.x >> 4;

    // coalesced load of the transposed-source tile rows (j0+ty, i0+tx)
    tT[ty][tx] = Anorm[(size_t)(j0 + ty) * NSAMP + i0 + tx]
               +  Bias[(size_t)(j0 + ty) * NSAMP + i0 + tx];
    __syncthreads();

    const int i = i0 + ty, j = j0 + tx;
    const float sv  = 1.0f / (1.0f + expf(-scalar_a[0]));
    const float fwd = Anorm[(size_t)i * NSAMP + j] + Bias[(size_t)i * NSAMP + j];
    const float bwd = tT[tx][ty];
    const float z   = sv * fwd + (1.0f - sv) * bwd;
    const float r   = 1.0f / (1.0f + expf(-z));
    out[(size_t)i * NSAMP + j] = (i == j) ? 0.0f : r;
}

// ---------------------------------------------------------------------------
extern "C" void kernel_launch(void* const* d_in, const int* in_sizes, int n_in,
                              void* d_out, int out_size, void* d_ws, size_t ws_size,
                              hipStream_t stream) {
    const float* x        = (const float*)d_in[0];
    const float* Wu_w     = (const float*)d_in[1];
    const float* Wu_b     = (const float*)d_in[2];
    const float* Wv_w     = (const float*)d_in[3];
    const float* Wv_b     = (const float*)d_in[4];
    const float* Wnorm_w  = (const float*)d_in[5];
    const float* scalar_a = (const float*)d_in[6];
    const float* AmatBias = (const float*)d_in[7];
    float* out = (float*)d_out;

    float* U  = (float*)d_ws;                          // 2048*720
    float* V  = U + (size_t)NSAMP * KDIM;              // 2048*720
    float* An = V + (size_t)NSAMP * KDIM;              // 2048*2048

    // K1: one block per sample
    proj_norm_kernel<<<NSAMP, 256, 0, stream>>>(x, Wu_w, Wu_b, Wv_w, Wv_b,
                                                Wnorm_w, U, V);

    // K2: 128 M-tiles * 32 N-groups = 4096 waves = 512 blocks of 8 waves
    gemm_wmma_kernel<<<(128 * 32) / 8, 256, 0, stream>>>(U, V, An);

    // K3: 128*128 16x16 tiles
    epilogue_kernel<<<128 * 128, 256, 0, stream>>>(An, AmatBias, scalar_a, out);
}